// TopKGate_20383914787047
// MI455X (gfx1250) — compile-verified
//
#include <hip/hip_runtime.h>
#include <stdint.h>

typedef __attribute__((ext_vector_type(2))) float v2f;
typedef __attribute__((ext_vector_type(8))) float v8f;

#define BATCH  8
#define NTOK   2048
#define DMODEL 1024
#define NEXP   64
#define CAP    80
#define DSZ    ((size_t)BATCH * NTOK * NEXP * CAP)   // 83,886,080 elements per dense output

// ---------------------------------------------------------------------------
// Kernel 1: zero-fill both dense outputs with wide stores (store-bound)
// ---------------------------------------------------------------------------
__global__ void zero_fill_kernel(float4* __restrict__ p, size_t n4) {
  size_t i = (size_t)blockIdx.x * blockDim.x + threadIdx.x;
  size_t stride = (size_t)gridDim.x * blockDim.x;
  float4 z; z.x = 0.f; z.y = 0.f; z.z = 0.f; z.w = 0.f;
  for (; i < n4; i += stride) p[i] = z;
}

// ---------------------------------------------------------------------------
// Kernel 2: gating GEMM (WMMA f32 16x16x4) + softmax + top-2 + random-second.
// One wave owns a 16-token row tile; 4 accumulators cover all 64 experts.
// ---------------------------------------------------------------------------
__global__ __launch_bounds__(256)
void gemm_gate_kernel(const float* __restrict__ x, const float* __restrict__ wg,
                      const float* __restrict__ noise,
                      uint32_t* __restrict__ packed, float* __restrict__ gate1,
                      float* __restrict__ gate2, float* __restrict__ sumP_part) {
  __shared__ float lds_sum[8 * NEXP];

  const int tid  = threadIdx.x;
  const int w    = tid >> 5;       // wave in block (8 waves)
  const int lane = tid & 31;
  const int half = lane >> 4;      // 0 or 1
  const int lq   = lane & 15;

  const int rt = blockIdx.x * 8 + w;       // global row tile 0..1023
  const int b  = rt >> 7;                  // 128 tiles per batch
  const int m0 = (rt & 127) << 4;          // token base within batch

  // A fragment (16x4 f32): lanes 0-15 hold K={0,1}; lanes 16-31 hold K={2,3}
  const float* aBase = x + ((size_t)b * NTOK + m0 + lq) * DMODEL + 2 * half;
  // B fragment (4x16 f32), w_gating is [K=1024][N=64] row-major
  const float* wBase = wg + (2 * half) * NEXP + lq;

  v8f acc0 = {}; v8f acc1 = {}; v8f acc2 = {}; v8f acc3 = {};

#pragma unroll 4
  for (int k0 = 0; k0 < DMODEL; k0 += 4) {
    v2f a = *(const v2f*)(aBase + k0);     // 8B-aligned
    const float* wp = wBase + k0 * NEXP;
    v2f b0; b0.x = wp[0];  b0.y = wp[NEXP + 0];
    v2f b1; b1.x = wp[16]; b1.y = wp[NEXP + 16];
    v2f b2; b2.x = wp[32]; b2.y = wp[NEXP + 32];
    v2f b3; b3.x = wp[48]; b3.y = wp[NEXP + 48];
    acc0 = __builtin_amdgcn_wmma_f32_16x16x4_f32(false, a, false, b0, (short)0, acc0, false, false);
    acc1 = __builtin_amdgcn_wmma_f32_16x16x4_f32(false, a, false, b1, (short)0, acc1, false, false);
    acc2 = __builtin_amdgcn_wmma_f32_16x16x4_f32(false, a, false, b2, (short)0, acc2, false, false);
    acc3 = __builtin_amdgcn_wmma_f32_16x16x4_f32(false, a, false, b3, (short)0, acc3, false, false);
  }

  // C/D layout: lane, reg r -> token m = r + 8*half, expert = 16*t + lq.
  // Per-token reductions stay inside each 16-lane half (xor offsets 1,2,4,8).
  float ps0 = 0.f, ps1 = 0.f, ps2 = 0.f, ps3 = 0.f;

  for (int r = 0; r < 8; ++r) {
    // stable softmax over 64 experts for token (m0 + r + 8*half)
    float m = fmaxf(fmaxf(acc0[r], acc1[r]), fmaxf(acc2[r], acc3[r]));
    for (int off = 1; off < 16; off <<= 1) m = fmaxf(m, __shfl_xor(m, off, 32));
    float e0 = __expf(acc0[r] - m);
    float e1 = __expf(acc1[r] - m);
    float e2 = __expf(acc2[r] - m);
    float e3 = __expf(acc3[r] - m);
    float s = e0 + e1 + e2 + e3;
    for (int off = 1; off < 16; off <<= 1) s += __shfl_xor(s, off, 32);
    float inv = 1.0f / s;
    float p0 = e0 * inv, p1 = e1 * inv, p2 = e2 * inv, p3 = e3 * inv;
    ps0 += p0; ps1 += p1; ps2 += p2; ps3 += p3;

    // lane-local top-2 over its 4 experts
    float v1 = p0, v2 = -1.f;
    int   i1 = lq, i2 = 0;
    { float p = p1; int e = 16 + lq;
      if (p > v1) { v2 = v1; i2 = i1; v1 = p; i1 = e; } else { v2 = p; i2 = e; } }
    { float p = p2; int e = 32 + lq;
      if (p > v1) { v2 = v1; i2 = i1; v1 = p; i1 = e; } else if (p > v2) { v2 = p; i2 = e; } }
    { float p = p3; int e = 48 + lq;
      if (p > v1) { v2 = v1; i2 = i1; v1 = p; i1 = e; } else if (p > v2) { v2 = p; i2 = e; } }

    // cross-lane merge of (top1, top2) pairs within the 16-lane half
    for (int off = 1; off < 16; off <<= 1) {
      float ov1 = __shfl_xor(v1, off, 32); int oi1 = __shfl_xor(i1, off, 32);
      float ov2 = __shfl_xor(v2, off, 32); int oi2 = __shfl_xor(i2, off, 32);
      if (ov1 > v1) {
        float c2v = (v1 > ov2) ? v1 : ov2;
        int   c2i = (v1 > ov2) ? i1 : oi2;
        v1 = ov1; i1 = oi1; v2 = c2v; i2 = c2i;
      } else if (ov1 > v2) {
        v2 = ov1; i2 = oi1;
      }
    }

    if (lq == 0) {
      int gidx = b * NTOK + m0 + r + 8 * half;
      float g1 = v1, g2 = v2;
      float denom = g1 + g2 + 1e-9f;
      g1 /= denom; g2 /= denom;
      // second policy 'random': keep iff noise < gate_2 / 0.2
      int keep = (noise[gidx] < g2 * 5.0f) ? 1 : 0;
      packed[gidx] = (uint32_t)i1 | ((uint32_t)i2 << 8) | ((uint32_t)keep << 16);
      gate1[gidx] = g1;
      gate2[gidx] = g2;
    }
  }

  // per-wave per-expert softmax sums (for load-balance loss), deterministic
  ps0 += __shfl_xor(ps0, 16, 32);
  ps1 += __shfl_xor(ps1, 16, 32);
  ps2 += __shfl_xor(ps2, 16, 32);
  ps3 += __shfl_xor(ps3, 16, 32);
  if (half == 0) {
    lds_sum[w * NEXP +  0 + lq] = ps0;
    lds_sum[w * NEXP + 16 + lq] = ps1;
    lds_sum[w * NEXP + 32 + lq] = ps2;
    lds_sum[w * NEXP + 48 + lq] = ps3;
  }
  __syncthreads();
  if (tid < NEXP) {
    float S = 0.f;
    for (int ww = 0; ww < 8; ++ww) S += lds_sum[ww * NEXP + tid];  // fixed order
    sumP_part[(size_t)blockIdx.x * NEXP + tid] = S;
  }
}

// ---------------------------------------------------------------------------
// Kernel 3: deterministic exclusive cumsum over tokens + scatter of non-zeros.
// One block per batch, one thread per expert (token order == jnp.cumsum order).
// ---------------------------------------------------------------------------
__global__ __launch_bounds__(64)
void scan_scatter_kernel(const uint32_t* __restrict__ packed,
                         const float* __restrict__ gate1,
                         const float* __restrict__ gate2,
                         float* __restrict__ out, int* __restrict__ c1_out) {
  const int b = blockIdx.x;
  const int e = threadIdx.x;
  const uint32_t* pk = packed + (size_t)b * NTOK;

  // pass 1: top-1 routing positions
  int c1 = 0;
  for (int i = 0; i < NTOK; ++i) {
    uint32_t p = pk[i];
    if ((int)(p & 0xffu) == e) {
      if (c1 < CAP) {
        float g = gate1[(size_t)b * NTOK + i];
        if (g > 0.f) {
          size_t base = (((size_t)b * NTOK + i) * NEXP + e) * CAP + c1;
          out[base]       = 1.0f;   // dispatch
          out[DSZ + base] = g;      // combine
        }
      }
      c1++;
    }
  }
  c1_out[b * NEXP + e] = c1;        // pre-capacity count (for loss density)

  // pass 2: top-2 positions start after min(c1, CAP) (mask_1_count)
  int base2 = (c1 < CAP) ? c1 : CAP;
  int c2 = 0;
  for (int i = 0; i < NTOK; ++i) {
    uint32_t p = pk[i];
    if ((int)((p >> 8) & 0xffu) == e && (p & 0x10000u)) {
      int pos = base2 + c2;
      if (pos < CAP) {
        float g = gate2[(size_t)b * NTOK + i];
        if (g > 0.f) {
          size_t base = (((size_t)b * NTOK + i) * NEXP + e) * CAP + pos;
          out[base]       = 1.0f;
          out[DSZ + base] = g;
        }
      }
      c2++;
    }
  }
}

// ---------------------------------------------------------------------------
// Kernel 4: load-balance loss, fixed-order reduction
// loss = mean_{b,e}( meanP * density ) * e^2 = (sum_{b,e} term) * e / b
// ---------------------------------------------------------------------------
__global__ __launch_bounds__(64)
void loss_kernel(const float* __restrict__ sumP_part, const int* __restrict__ c1,
                 float* __restrict__ out) {
  __shared__ float sh[NEXP];
  int e = threadIdx.x;
  float term = 0.f;
  for (int b = 0; b < BATCH; ++b) {
    float S = 0.f;
    for (int k = 0; k < 16; ++k) S += sumP_part[((size_t)b * 16 + k) * NEXP + e];
    term += (S / (float)NTOK) * ((float)c1[b * NEXP + e] / (float)NTOK);
  }
  sh[e] = term;
  __syncthreads();
  if (e == 0) {
    float tot = 0.f;
    for (int k = 0; k < NEXP; ++k) tot += sh[k];
    out[2 * DSZ] = tot * ((float)(NEXP * NEXP) / (float)(BATCH * NEXP)); // * 8
  }
}

// ---------------------------------------------------------------------------
extern "C" void kernel_launch(void* const* d_in, const int* in_sizes, int n_in,
                              void* d_out, int out_size, void* d_ws, size_t ws_size,
                              hipStream_t stream) {
  (void)in_sizes; (void)n_in; (void)out_size; (void)ws_size;
  const float* x     = (const float*)d_in[0];
  const float* wgate = (const float*)d_in[1];
  const float* noise = (const float*)d_in[2];
  float* out = (float*)d_out;

  char* ws = (char*)d_ws;
  uint32_t* packed    = (uint32_t*)(ws);                       // 16384 * 4
  float*    gate1     = (float*)   (ws + 1 * 65536);
  float*    gate2     = (float*)   (ws + 2 * 65536);
  float*    sumP_part = (float*)   (ws + 3 * 65536);           // 128 * 64 * 4
  int*      c1        = (int*)     (ws + 3 * 65536 + 32768);   // 8 * 64 * 4

  size_t n4 = (2 * DSZ) / 4;
  zero_fill_kernel  <<<16384, 256, 0, stream>>>((float4*)out, n4);
  gemm_gate_kernel  <<<128,   256, 0, stream>>>(x, wgate, noise, packed, gate1, gate2, sumP_part);
  scan_scatter_kernel<<<BATCH, 64, 0, stream>>>(packed, gate1, gate2, out, c1);
  loss_kernel       <<<1,      64, 0, stream>>>(sumP_part, c1, out);
}